// NSVQ_39556648796218
// MI455X (gfx1250) — compile-verified
//
#include <hip/hip_runtime.h>

typedef __attribute__((ext_vector_type(2))) float v2f;
typedef __attribute__((ext_vector_type(4))) float v4f;
typedef __attribute__((ext_vector_type(8))) float v8f;

#define EMBEDDING_DIM 64
#define NUM_STAGES 4
#define N_TOKENS 1048576

// Output is N_TOKENS x 64 f32 = 16,777,216 float4s = 256 MB of pure stores.
#define TOTAL_F4 (N_TOKENS * (EMBEDDING_DIM / 4))
#define TPB 256
#define BLOCKS 4096
#define ITERS (TOTAL_F4 / (TPB * BLOCKS)) // 16

// ---------------------------------------------------------------------------
// Stage-sum via V_WMMA_F32_16X16X4_F32 (exact f32):
//   D = A(16x4) * B(4x16) + 0, with B = all-ones, so D[m,n] = sum_k A[m,k].
//   A[m][k] = codebooks[k][16*t + m]  => D[m,*] = s[16*t + m].
// A-layout (ISA 7.12.2, 32-bit 16x4): lane 0-15: M=lane, VGPR0=K0, VGPR1=K1;
//                                     lane 16-31: M=lane-16, VGPR0=K2, VGPR1=K3.
// B = 1.0 everywhere => its lane layout cannot affect the result.
// D-layout: VGPR r: lanes 0-15 hold M=r, lanes 16-31 hold M=r+8 (N = lane&15).
// ---------------------------------------------------------------------------
__global__ __launch_bounds__(32) void nsvq_sum_wmma(const float* __restrict__ cb,
                                                    float* __restrict__ s) {
  const int lane = threadIdx.x;      // wave32, one full wave -> EXEC all ones
  const int m    = lane & 15;
  const int kb   = (lane >> 4) << 1; // 0 for lanes 0-15, 2 for lanes 16-31

  v2f b;
  b.x = 1.0f;
  b.y = 1.0f;

#pragma unroll
  for (int t = 0; t < NUM_STAGES; ++t) {
    const int d = t * 16 + m;
    v2f a;
    a.x = cb[(kb + 0) * EMBEDDING_DIM + d]; // A[m][kb]
    a.y = cb[(kb + 1) * EMBEDDING_DIM + d]; // A[m][kb+1]
    v8f c = {};
    c = __builtin_amdgcn_wmma_f32_16x16x4_f32(
        /*neg_a=*/false, a, /*neg_b=*/false, b,
        /*c_mod=*/(short)0, c, /*reuse_a=*/false, /*reuse_b=*/false);
    // Every column n holds the same value; lane 0 carries s[16t+0..7] in c[0..7],
    // lane 16 carries s[16t+8..15].
    if (m == 0) {
      const int base = t * 16 + ((lane >> 4) << 3);
#pragma unroll
      for (int r = 0; r < 8; ++r) s[base + r] = c[r];
    }
  }
}

// ---------------------------------------------------------------------------
// Broadcast: every row of the output is the same 64-float vector.
// Each thread owns one fixed float4 column slice (d4 = tid & 15 stays constant
// because the stride TPB*BLOCKS is a multiple of 16) and streams 16 NT b128
// stores. 256 MB / 23.3 TB/s ~= 11 us floor; this kernel is store-bound only.
// ---------------------------------------------------------------------------
__global__ __launch_bounds__(TPB) void nsvq_bcast(const float* __restrict__ s,
                                                  float* __restrict__ out) {
  const int tid = blockIdx.x * TPB + threadIdx.x;
  const v4f c = ((const v4f*)s)[tid & 15]; // 256B table, L0/L2 broadcast hit
  v4f* o = (v4f*)out;
  size_t j = (size_t)tid;
#pragma unroll
  for (int i = 0; i < ITERS; ++i) {
    __builtin_nontemporal_store(c, o + j); // global_store_b128, NT hint
    j += (size_t)(TPB * BLOCKS);
  }
}

// Fallback if workspace is unusable: compute the sum inline with the exact
// reference association order (((c0+c1)+c2)+c3) — 4 cached 1KB reads total.
__global__ __launch_bounds__(TPB) void nsvq_bcast_inline(const float* __restrict__ cb,
                                                         float* __restrict__ out) {
  const int tid = blockIdx.x * TPB + threadIdx.x;
  const int d4 = tid & 15;
  const v4f* cb4 = (const v4f*)cb;
  v4f c = cb4[0 * 16 + d4];
  c = c + cb4[1 * 16 + d4];
  c = c + cb4[2 * 16 + d4];
  c = c + cb4[3 * 16 + d4];
  v4f* o = (v4f*)out;
  size_t j = (size_t)tid;
#pragma unroll
  for (int i = 0; i < ITERS; ++i) {
    __builtin_nontemporal_store(c, o + j);
    j += (size_t)(TPB * BLOCKS);
  }
}

extern "C" void kernel_launch(void* const* d_in, const int* in_sizes, int n_in,
                              void* d_out, int out_size, void* d_ws, size_t ws_size,
                              hipStream_t stream) {
  // d_in[0] = input_data (f32, 1048576x64)  -- mathematically unused: the
  //           reference's argmin is over a size-1 axis, so indices are all 0
  //           and the output is input-independent.
  // d_in[1] = codebooks (f32, 4x64)
  const float* cb = (const float*)d_in[1];
  float* out = (float*)d_out;

  if (d_ws != nullptr && ws_size >= EMBEDDING_DIM * sizeof(float)) {
    float* s = (float*)d_ws;
    nsvq_sum_wmma<<<1, 32, 0, stream>>>(cb, s);          // one wave, 4 WMMAs
    nsvq_bcast<<<BLOCKS, TPB, 0, stream>>>(s, out);      // 256 MB NT stores
  } else {
    nsvq_bcast_inline<<<BLOCKS, TPB, 0, stream>>>(cb, out);
  }
  (void)d_in; (void)in_sizes; (void)n_in; (void)out_size;
}